// LF5DGrid_70471823393088
// MI455X (gfx1250) — compile-verified
//
#include <hip/hip_runtime.h>
#include <hip/hip_bf16.h>
#include <stdint.h>

typedef __attribute__((ext_vector_type(2))) float v2f;
typedef __attribute__((ext_vector_type(8))) float v8f;

#define GP (1 << 20)   // 16^5 cells
#define GC 32          // channels
#define GDIM 16

// ---------------------------------------------------------------------------
// Pass 1: transpose grid (C=32, P) -> grid_t (P, 32) so that the 32 channels
// of one cell form a single contiguous 128-byte cache line.
// ---------------------------------------------------------------------------
__global__ __launch_bounds__(256) void lf5d_transpose(
    const float* __restrict__ g, float* __restrict__ gt) {
  __shared__ float tile[32][65];               // stride 65 -> bank-conflict free
  const int tid = threadIdx.x;
  const int p0  = blockIdx.x * 64;
  const int pl  = tid & 63;                    // cell within tile
  const int cb  = tid >> 6;                    // 0..3
#pragma unroll
  for (int c = cb; c < 32; c += 4)             // coalesced along P
    tile[c][pl] = g[(size_t)c * GP + p0 + pl];
  __syncthreads();
  const int c2 = tid & 31;
  const int qb = tid >> 5;                     // 0..7
#pragma unroll
  for (int q = qb; q < 64; q += 8)             // coalesced along C
    gt[(size_t)(p0 + q) * GC + c2] = tile[c2][q];
}

// ---------------------------------------------------------------------------
// Pass 2: interpolation. One wave handles 128 consecutive rays.
//  - wave async-stages its 128 rays (2560B = exactly 5 b128 ops/lane) to LDS
//  - per ray: lane = (half<<4)|ch ; half selects corner pair {0,1} / {2,3}
//  - corner reduction (K=4 contraction) done with V_WMMA_F32_16X16X4_F32
//  - store: every lane holds channel `lane` -> one coalesced 128B store
//  - ray-loop trip count is forced into an SGPR (readfirstlane) so loop
//    control is pure SALU and EXEC is provably all-1s at every WMMA
// ---------------------------------------------------------------------------
template <int CELLS, int CHANS>
__global__ __launch_bounds__(256) void lf5d_interp(
    const float* __restrict__ ray, const float* __restrict__ grid,
    const float* __restrict__ rmin, const float* __restrict__ rmax,
    float* __restrict__ out, int n) {
  __shared__ float stage[8 * 640];             // 8 waves x 128 rays x 5 floats
  const int tid  = threadIdx.x;
  const int lane = tid & 31;
  const int wv   = tid >> 5;
  float* myStage = &stage[wv * 640];

  const long wgid    = (long)blockIdx.x * 8 + wv;
  const long rayBase = wgid * 128;

  // ---- async-stage this wave's rays into LDS (ASYNCcnt path) -------------
  {
    const long totalBytes = (long)n * 5L * 4L;
    const long maxOfs     = (totalBytes - 16) & ~15L;
    const unsigned ldsBase = (unsigned)(uintptr_t)myStage;
#pragma unroll
    for (int i = 0; i < 5; ++i) {
      long gofs = rayBase * 20 + (long)i * 512 + (long)lane * 16;
      if (gofs > maxOfs) gofs = maxOfs;        // clamp tail (keeps EXEC full)
      if (gofs < 0)      gofs = 0;
      const unsigned lofs = ldsBase + (unsigned)(i * 512 + lane * 16);
      const unsigned long long gaddr =
          (unsigned long long)(uintptr_t)ray + (unsigned long long)gofs;
      asm volatile("global_load_async_to_lds_b128 %0, %1, off"
                   :: "v"(lofs), "v"(gaddr) : "memory");
    }
    asm volatile("s_wait_asynccnt 0" ::: "memory");
  }

  // ---- per-dim normalization (all dims are 16 wide) ----------------------
  float mn[5], sc[5];
#pragma unroll
  for (int d = 0; d < 5; ++d) {
    mn[d] = rmin[d];
    sc[d] = (float)(GDIM - 1) / (rmax[d] - mn[d]);
  }

  const int half = lane >> 4;                  // corner-pair select
  const int ch   = lane & 15;                  // channel within 16-chunk
  // loop-invariant channel offsets (elements)
  const unsigned co0 = (unsigned)ch * (unsigned)CHANS;
  const unsigned co1 = (unsigned)(ch + 16) * (unsigned)CHANS;

  // scalar (SGPR) trip count: loop control is SALU-only, EXEC stays all-1s
  long rem = (long)n - rayBase;
  int cnt = rem < 0 ? 0 : (rem > 128 ? 128 : (int)rem);
  cnt = __builtin_amdgcn_readfirstlane(cnt);

  float* outp = out + rayBase * GC + lane;     // advances by 128B per ray

  for (int j = 0; j < cnt; ++j) {              // scalar loop bound
    const float* rp = &myStage[j * 5];

    float w[5]; int b[5];
#pragma unroll
    for (int d = 0; d < 5; ++d) {
      float ind = (rp[d] - mn[d]) * sc[d];
      float fl  = floorf(ind);
      b[d] = (int)fl;
      w[d] = ind - fl;
    }
    const float q = (1.f - w[2]) * (1.f - w[3]) * (1.f - w[4]);
    const bool vt = (b[2] >= 0) & (b[2] < GDIM) & (b[3] >= 0) & (b[3] < GDIM) &
                    (b[4] >= 0) & (b[4] < GDIM);
    const int tail = b[2] * 256 + b[3] * 16 + b[4];

    float wk[2]; unsigned off[2];
#pragma unroll
    for (int t = 0; t < 2; ++t) {
      const int k  = half * 2 + t;             // corner id 0..3
      const int o0 = k & 1, o1 = (k >> 1) & 1; // CORNER_OFFSETS row k
      const int c0 = b[0] + o0, c1 = b[1] + o1;
      const bool v = vt & (c0 >= 0) & (c0 < GDIM) & (c1 >= 0) & (c1 < GDIM);
      const float ww = (o0 ? w[0] : 1.f - w[0]) * (o1 ? w[1] : 1.f - w[1]) * q;
      wk[t] = v ? ww : 0.f;
      int lin = c0 * 65536 + c1 * 4096 + tail;       // int32, like reference
      lin = lin < 0 ? 0 : lin;
      lin = lin > (GP - 1) ? (GP - 1) : lin;         // med3-style clamp
      off[t] = (unsigned)lin * (unsigned)CELLS;      // CELLS is 32 or 1: shift
    }

    // A-matrix 16x4 f32 layout: lanes 0-15 VGPRs {K0,K1}; lanes 16-31 {K2,K3}
    v2f A; A[0] = wk[0]; A[1] = wk[1];
    // B-matrix 4x16 f32: lanes 0-15 N=col, VGPRs {K0,K1}; lanes 16-31 {K2,K3}
    v2f B0; B0[0] = grid[off[0] + co0]; B0[1] = grid[off[1] + co0];
    v2f B1; B1[0] = grid[off[0] + co1]; B1[1] = grid[off[1] + co1];

    const v8f z = {0.f, 0.f, 0.f, 0.f, 0.f, 0.f, 0.f, 0.f};
    // D[m,c] = sum_k w_k * v_k[c]  (rows identical since A rows identical)
    v8f d0 = __builtin_amdgcn_wmma_f32_16x16x4_f32(
        false, A, false, B0, (short)0, z, false, false);
    v8f d1 = __builtin_amdgcn_wmma_f32_16x16x4_f32(
        false, A, false, B1, (short)0, z, false, false);

    // lane l: d[0] holds column (l&15); halves map so channel == lane exactly
    *outp = half ? d1[0] : d0[0];
    outp += GC;
  }
}

// ---------------------------------------------------------------------------
extern "C" void kernel_launch(void* const* d_in, const int* in_sizes, int n_in,
                              void* d_out, int out_size, void* d_ws,
                              size_t ws_size, hipStream_t stream) {
  const float* ray  = (const float*)d_in[0];
  const float* grid = (const float*)d_in[1];
  const float* rmin = (const float*)d_in[2];
  const float* rmax = (const float*)d_in[3];
  float* out = (float*)d_out;

  const int n = in_sizes[0] / 5;
  const int blocks = (n + 1023) / 1024;        // 1024 rays per block (8 waves)
  const size_t need = (size_t)GP * GC * sizeof(float);   // 128 MB

  if (ws_size >= need) {
    float* gt = (float*)d_ws;
    lf5d_transpose<<<GP / 64, 256, 0, stream>>>(grid, gt);
    lf5d_interp<GC, 1><<<blocks, 256, 0, stream>>>(ray, gt, rmin, rmax, out, n);
  } else {
    lf5d_interp<1, GP><<<blocks, 256, 0, stream>>>(ray, grid, rmin, rmax, out, n);
  }
}